// FrequencyLayer_81097572483428
// MI455X (gfx1250) — compile-verified
//
#include <hip/hip_runtime.h>

typedef __attribute__((ext_vector_type(2))) float v2f;
typedef __attribute__((ext_vector_type(8))) float v8f;

// 8x8 DCT-II basis D[k][i] = s(k) * cos(pi*(2i+1)*k/16), s(0)=sqrt(1/8), s(k>0)=0.5
__constant__ float DCT8[64] = {
    0.35355339f,  0.35355339f,  0.35355339f,  0.35355339f,  0.35355339f,  0.35355339f,  0.35355339f,  0.35355339f,
    0.49039264f,  0.41573481f,  0.27778512f,  0.09754516f, -0.09754516f, -0.27778512f, -0.41573481f, -0.49039264f,
    0.46193977f,  0.19134172f, -0.19134172f, -0.46193977f, -0.46193977f, -0.19134172f,  0.19134172f,  0.46193977f,
    0.41573481f, -0.09754516f, -0.49039264f, -0.27778512f,  0.27778512f,  0.49039264f,  0.09754516f, -0.41573481f,
    0.35355339f, -0.35355339f, -0.35355339f,  0.35355339f,  0.35355339f, -0.35355339f, -0.35355339f,  0.35355339f,
    0.27778512f, -0.49039264f,  0.09754516f,  0.41573481f, -0.41573481f, -0.09754516f,  0.49039264f, -0.27778512f,
    0.19134172f, -0.46193977f,  0.46193977f, -0.19134172f, -0.19134172f,  0.46193977f, -0.46193977f,  0.19134172f,
    0.09754516f, -0.27778512f,  0.41573481f, -0.49039264f,  0.49039264f, -0.41573481f,  0.27778512f, -0.09754516f
};

// Block-diagonal 16x16 basis diag(D, D): zero when row/col are in different 8-blocks.
__device__ __forceinline__ float blockD(int r, int c) {
    float v = DCT8[((r & 7) << 3) | (c & 7)];
    return ((r ^ c) & 8) ? 0.0f : v;
}

// One wave handles one 16x16 pixel tile = 2x2 patches of 8x8.
// Z = diag(D,D) * X * diag(D,D)^T computed as two chains of 4x v_wmma_f32_16x16x4_f32.
__global__ __launch_bounds__(256)
void freq_dct_kernel(const float* __restrict__ x,
                     float* __restrict__ out_low,
                     float* __restrict__ out_high) {
    const int lane = threadIdx.x & 31;
    const int wid  = threadIdx.x >> 5;
    const int tile = blockIdx.x * 8 + wid;          // 0 .. 196607

    const int tw = tile & 31;                       // tile col (of 32)
    const int th = (tile >> 5) & 31;                // tile row (of 32)
    const int bc = tile >> 10;                      // b*3 + c
    const int c  = bc % 3;
    const int b  = bc / 3;

    const int m  = lane & 15;   // M index (A rows) and N index (B/C cols)
    const int hi = lane >> 4;   // lane half selects K offset (+2) per documented 16x4 layout

    const float* src = x + ((size_t)bc * 512 + (size_t)th * 16) * 512 + (size_t)tw * 16;

    // ---- Product 1: Y = Dblk * X ----
    v8f acc = {};
#pragma unroll
    for (int kc = 0; kc < 4; ++kc) {
        const int r0 = kc * 4 + 2 * hi;             // K indices r0, r0+1 for this VGPR pair
        v2f a, bb;
        a.x  = blockD(m, r0);                       // A[m][r0]
        a.y  = blockD(m, r0 + 1);                   // A[m][r0+1]
        bb.x = src[(size_t)r0 * 512 + m];           // B[r0][m]   = X[r0][m]
        bb.y = src[(size_t)(r0 + 1) * 512 + m];     // B[r0+1][m]
        acc = __builtin_amdgcn_wmma_f32_16x16x4_f32(
            /*neg_a=*/false, a, /*neg_b=*/false, bb,
            /*c_mod=*/(short)0, acc, /*reuse_a=*/false, /*reuse_b=*/false);
    }

    // ---- Relayout Y through LDS (C layout -> A layout), 16x17 pad vs bank conflicts ----
    __shared__ float lds[8][16 * 17];
    float* Y = &lds[wid][0];
#pragma unroll
    for (int v = 0; v < 8; ++v)
        Y[(v + 8 * hi) * 17 + m] = acc[v];          // Z row M = v + 8*hi, col N = m
    asm volatile("s_wait_dscnt 0" ::: "memory");    // wave-local DS ordering

    // ---- Product 2: Z = Y * Dblk^T ----
    v8f z = {};
#pragma unroll
    for (int kc = 0; kc < 4; ++kc) {
        const int k0 = kc * 4 + 2 * hi;
        v2f a, bb;
        a.x  = Y[m * 17 + k0];                      // A[m][k0] = Y[m][k0]
        a.y  = Y[m * 17 + k0 + 1];
        bb.x = blockD(m, k0);                       // B[k][n] = Dblk^T[k][n] = Dblk[n][k], n=m
        bb.y = blockD(m, k0 + 1);
        z = __builtin_amdgcn_wmma_f32_16x16x4_f32(
            false, a, false, bb, (short)0, z, false, false);
    }

    // ---- Z back to LDS, then 8 lanes emit float4 per (patch, low/high) ----
#pragma unroll
    for (int v = 0; v < 8; ++v)
        Y[(v + 8 * hi) * 17 + m] = z[v];
    asm volatile("s_wait_dscnt 0" ::: "memory");

    if (lane < 8) {
        const int p   = lane >> 1;                  // patch 0..3 within tile
        const int pr  = p >> 1, pc = p & 1;
        const int sel = lane & 1;                   // 0 = low, 1 = high
        const int ph  = th * 2 + pr;
        const int pw  = tw * 2 + pc;
        const size_t obase = ((((size_t)b * 64 + ph) * 64 + pw) * 3 + c) * 4;
        const int rb = pr * 8, cb = pc * 8;
        float4 vo;
        if (sel == 0) {
            vo.x = Y[(rb + 0) * 17 + cb + 0];
            vo.y = Y[(rb + 0) * 17 + cb + 1];
            vo.z = Y[(rb + 1) * 17 + cb + 0];
            vo.w = Y[(rb + 1) * 17 + cb + 1];
            *(float4*)(out_low + obase) = vo;
        } else {                                    // high[a][b] = d[7-a][7-b]
            vo.x = Y[(rb + 7) * 17 + cb + 7];
            vo.y = Y[(rb + 7) * 17 + cb + 6];
            vo.z = Y[(rb + 6) * 17 + cb + 7];
            vo.w = Y[(rb + 6) * 17 + cb + 6];
            *(float4*)(out_high + obase) = vo;
        }
    }
}

extern "C" void kernel_launch(void* const* d_in, const int* in_sizes, int n_in,
                              void* d_out, int out_size, void* d_ws, size_t ws_size,
                              hipStream_t stream) {
    (void)in_sizes; (void)n_in; (void)out_size; (void)d_ws; (void)ws_size;
    const float* x = (const float*)d_in[0];
    float* out = (float*)d_out;
    const size_t half = (size_t)64 * 64 * 64 * 3 * 4;   // 3,145,728 floats per output
    const int totalTiles = 64 * 3 * 32 * 32;            // 196,608 tiles, 8 waves/block
    dim3 grid(totalTiles / 8), block(256);
    freq_dct_kernel<<<grid, block, 0, stream>>>(x, out, out + half);
}